// Encoder_base_18305150616329
// MI455X (gfx1250) — compile-verified
//
#include <hip/hip_runtime.h>
#include <hip/hip_bf16.h>

typedef float v2f __attribute__((ext_vector_type(2)));
typedef float v8f __attribute__((ext_vector_type(8)));

#define BN_EPS 1e-5f

// ---------------------------------------------------------------------------
// agg = h   (float4 coalesced copy; GIN self term)
// ---------------------------------------------------------------------------
__global__ void __launch_bounds__(256) copy4_kernel(const float* __restrict__ in,
                                                    float* __restrict__ out, int n4) {
    int t = blockIdx.x * 256 + threadIdx.x;
    if (t >= n4) return;
    *(float4*)(out + (size_t)t * 4) = *(const float4*)(in + (size_t)t * 4);
}

// ---------------------------------------------------------------------------
// agg[dst] += h[src] over edges. 16 threads per edge, float4 each -> the 16
// threads cover one edge's 64 floats (256B contiguous) for both the gather
// and the atomic destinations. fp32 relaxed agent-scope atomics lower to
// native global_atomic_add_f32.
// ---------------------------------------------------------------------------
__global__ void __launch_bounds__(256) scatter_add_kernel(const float* __restrict__ h,
                                                          const int* __restrict__ src,
                                                          const int* __restrict__ dst,
                                                          float* __restrict__ agg,
                                                          int nEdges) {
    int t = blockIdx.x * 256 + threadIdx.x;
    int e = t >> 4;
    if (e >= nEdges) return;
    int c = (t & 15) << 2;
    int s = src[e];
    int d = dst[e];
    float4 v = *(const float4*)(h + (size_t)s * 64 + c);
    float* p = agg + (size_t)d * 64 + c;
    __hip_atomic_fetch_add(p + 0, v.x, __ATOMIC_RELAXED, __HIP_MEMORY_SCOPE_AGENT);
    __hip_atomic_fetch_add(p + 1, v.y, __ATOMIC_RELAXED, __HIP_MEMORY_SCOPE_AGENT);
    __hip_atomic_fetch_add(p + 2, v.z, __ATOMIC_RELAXED, __HIP_MEMORY_SCOPE_AGENT);
    __hip_atomic_fetch_add(p + 3, v.w, __ATOMIC_RELAXED, __HIP_MEMORY_SCOPE_AGENT);
}

// ---------------------------------------------------------------------------
// out[N,64] = relu(A[N,64] @ W[64,64] + bias), fp32 WMMA (V_WMMA_F32_16X16X4_F32).
// Block = 128 threads = 4 waves. Each wave: one 16-row tile x 64 output cols,
// 4 v8f accumulators, 16 k-steps x 4 n-tiles = 64 WMMA ops.
//
// W is staged in LDS pair-interleaved: Wp[p][c] = (W[2p][c], W[2p+1][c]) with
// a pair-row pitch of 80 float2 (640B). Each B fragment is then ONE aligned
// 8-byte ds_load_b64 straight into an even VGPR pair (no marshaling movs),
// and pitch 80 gives a 32-bank shift between consecutive pair-rows, so the
// two lane halves (pair 2k vs 2k+1) hit disjoint bank halves: conflict-free.
//
// fp32 fragment layouts (ISA 7.12.2), lane = 16*half + m:
//   A 16x4 : a = { A[m][4k+2*half+0], A[m][4k+2*half+1] }  (one global b64)
//   B 4x16 : b = Wp[2k+half][col],  col = nt*16+m          (one ds b64)
//   C 16x16: acc[r] = C[half*8 + r][col]
// ---------------------------------------------------------------------------
#define WP_PITCH 80   // float2 elements per pair-row

__global__ void __launch_bounds__(128) gemm64_relu_kernel(const float* __restrict__ A,
                                                          const float* __restrict__ W,
                                                          const float* __restrict__ bias,
                                                          float* __restrict__ out,
                                                          int nRows) {
    __shared__ float Wp[32 * WP_PITCH * 2];   // 32 pair-rows x 80 float2
    __shared__ float bs[64];
    for (int i = threadIdx.x; i < 64 * 64; i += 128) {
        int r = i >> 6, c = i & 63;
        int p = r >> 1, comp = r & 1;
        Wp[(p * WP_PITCH + c) * 2 + comp] = W[i];
    }
    if (threadIdx.x < 64) bs[threadIdx.x] = bias[threadIdx.x];
    __syncthreads();

    int wave = threadIdx.x >> 5;
    int lane = threadIdx.x & 31;
    int rowTiles = (nRows + 15) >> 4;
    int tile = blockIdx.x * 4 + wave;   // wave-uniform
    if (tile >= rowTiles) return;       // whole wave exits -> EXEC stays all-1s

    int m = lane & 15;
    int half = lane >> 4;
    int base = tile * 16;
    int arow = base + m;
    if (arow > nRows - 1) arow = nRows - 1;   // clamp (no divergence)
    const float* ap = A + (size_t)arow * 64 + 2 * half;

    v8f acc0 = {}, acc1 = {}, acc2 = {}, acc3 = {};
    for (int k = 0; k < 16; ++k) {
        v2f a = *(const v2f*)(ap + 4 * k);                       // global_load_b64
        const float* wrow = &Wp[((2 * k + half) * WP_PITCH + m) * 2];
        v2f b0 = *(const v2f*)(wrow + 0);                        // ds_load_b64
        v2f b1 = *(const v2f*)(wrow + 32);
        v2f b2 = *(const v2f*)(wrow + 64);
        v2f b3 = *(const v2f*)(wrow + 96);
        acc0 = __builtin_amdgcn_wmma_f32_16x16x4_f32(false, a, false, b0, (short)0, acc0, false, false);
        acc1 = __builtin_amdgcn_wmma_f32_16x16x4_f32(false, a, false, b1, (short)0, acc1, false, false);
        acc2 = __builtin_amdgcn_wmma_f32_16x16x4_f32(false, a, false, b2, (short)0, acc2, false, false);
        acc3 = __builtin_amdgcn_wmma_f32_16x16x4_f32(false, a, false, b3, (short)0, acc3, false, false);
    }

    float bc0 = bs[m], bc1 = bs[16 + m], bc2 = bs[32 + m], bc3 = bs[48 + m];
    int rowOff = base + half * 8;
#pragma unroll
    for (int r = 0; r < 8; ++r) {
        int row = rowOff + r;
        if (row < nRows) {
            float* op = out + (size_t)row * 64 + m;
            float v0 = acc0[r] + bc0; op[0]  = v0 > 0.f ? v0 : 0.f;
            float v1 = acc1[r] + bc1; op[16] = v1 > 0.f ? v1 : 0.f;
            float v2 = acc2[r] + bc2; op[32] = v2 > 0.f ? v2 : 0.f;
            float v3 = acc3[r] + bc3; op[48] = v3 > 0.f ? v3 : 0.f;
        }
    }
}

// ---------------------------------------------------------------------------
// Per-feature sum / sumsq over the node dimension (BN training stats).
// stats[0:64] = sum, stats[64:128] = sumsq (must be zeroed before launch).
// ---------------------------------------------------------------------------
__global__ void __launch_bounds__(256) bn_stats_kernel(const float* __restrict__ r,
                                                       float* __restrict__ stats,
                                                       int nRows) {
    __shared__ float shs[256];
    __shared__ float shq[256];
    int col = threadIdx.x & 63;
    int grp = threadIdx.x >> 6;
    float s = 0.f, q = 0.f;
    for (int row = blockIdx.x * 4 + grp; row < nRows; row += gridDim.x * 4) {
        float v = r[(size_t)row * 64 + col];
        s += v;
        q += v * v;
    }
    shs[threadIdx.x] = s;
    shq[threadIdx.x] = q;
    __syncthreads();
    if (grp == 0) {
        s = shs[col] + shs[64 + col] + shs[128 + col] + shs[192 + col];
        q = shq[col] + shq[64 + col] + shq[128 + col] + shq[192 + col];
        __hip_atomic_fetch_add(&stats[col], s, __ATOMIC_RELAXED, __HIP_MEMORY_SCOPE_AGENT);
        __hip_atomic_fetch_add(&stats[64 + col], q, __ATOMIC_RELAXED, __HIP_MEMORY_SCOPE_AGENT);
    }
}

// scale/shift: ss[c] = g*rsqrt(var+eps), ss[64+c] = b - mean*scale
__global__ void bn_finalize_kernel(const float* __restrict__ stats,
                                   const float* __restrict__ g,
                                   const float* __restrict__ b,
                                   float* __restrict__ ss, float invN) {
    int c = threadIdx.x;
    if (c >= 64) return;
    float mean = stats[c] * invN;
    float var  = stats[64 + c] * invN - mean * mean;
    float sc   = g[c] * rsqrtf(var + BN_EPS);
    ss[c]      = sc;
    ss[64 + c] = b[c] - mean * sc;
}

// out = r * scale[col] + shift[col]   (float4 vectorized)
__global__ void __launch_bounds__(256) bn_apply_kernel(const float* __restrict__ r,
                                                       const float* __restrict__ ss,
                                                       float* __restrict__ out, int n4) {
    int t = blockIdx.x * 256 + threadIdx.x;
    if (t >= n4) return;
    int i = t << 2;
    int c = i & 63;
    float4 v = *(const float4*)(r + i);
    float4 o;
    o.x = v.x * ss[c + 0] + ss[64 + c + 0];
    o.y = v.y * ss[c + 1] + ss[64 + c + 1];
    o.z = v.z * ss[c + 2] + ss[64 + c + 2];
    o.w = v.w * ss[c + 3] + ss[64 + c + 3];
    *(float4*)(out + i) = o;
}

// ---------------------------------------------------------------------------
extern "C" void kernel_launch(void* const* d_in, const int* in_sizes, int n_in,
                              void* d_out, int out_size, void* d_ws, size_t ws_size,
                              hipStream_t stream) {
    const int D = 64, L = 3;
    const int N = in_sizes[0] / D;
    const int E = in_sizes[1] / 2;
    const int ND = N * D;
    const int n4 = ND / 4;

    const float* x    = (const float*)d_in[0];
    const int*   ei   = (const int*)d_in[1];
    const int*   src  = ei;
    const int*   dst  = ei + E;
    const float* W1   = (const float*)d_in[3];
    const float* b1   = (const float*)d_in[4];
    const float* W2   = (const float*)d_in[5];
    const float* b2   = (const float*)d_in[6];
    const float* bn_g = (const float*)d_in[7];
    const float* bn_b = (const float*)d_in[8];
    const float* muW  = (const float*)d_in[9];
    const float* mub  = (const float*)d_in[10];
    const float* lvW  = (const float*)d_in[11];
    const float* lvb  = (const float*)d_in[12];
    const float* mug  = (const float*)d_in[13];
    const float* mubt = (const float*)d_in[14];
    const float* lvg  = (const float*)d_in[15];
    const float* lvbt = (const float*)d_in[16];

    float* ws    = (float*)d_ws;
    float* h     = ws;                  // [N,64] normalized features
    float* agg   = ws + (size_t)ND;     // [N,64] aggregate / relu(mlp)
    float* tmp   = ws + (size_t)2 * ND; // [N,64] hidden
    float* stats = ws + (size_t)3 * ND; // 128 floats: sum|sumsq
    float* ss    = stats + 128;         // 128 floats: scale|shift

    const float invN = 1.0f / (float)N;
    const int gridElem = (n4 + 255) / 256;
    const int gridScat = (E * 16 + 255) / 256;
    const int rowTiles = (N + 15) / 16;
    const int gridGemm = (rowTiles + 3) / 4;

    const float* hin = x;
    for (int l = 0; l < L; ++l) {
        // agg = hin + scatter_add(hin[src] -> dst)
        copy4_kernel<<<gridElem, 256, 0, stream>>>(hin, agg, n4);
        scatter_add_kernel<<<gridScat, 256, 0, stream>>>(hin, src, dst, agg, E);
        // tmp = relu(agg @ W1 + b1)
        gemm64_relu_kernel<<<gridGemm, 128, 0, stream>>>(agg, W1 + (size_t)l * D * D,
                                                         b1 + l * D, tmp, N);
        // agg = relu(tmp @ W2 + b2)
        gemm64_relu_kernel<<<gridGemm, 128, 0, stream>>>(tmp, W2 + (size_t)l * D * D,
                                                         b2 + l * D, agg, N);
        // BN over node dim -> h
        hipMemsetAsync(stats, 0, 128 * sizeof(float), stream);
        bn_stats_kernel<<<1024, 256, 0, stream>>>(agg, stats, N);
        bn_finalize_kernel<<<1, 64, 0, stream>>>(stats, bn_g + l * D, bn_b + l * D, ss, invN);
        bn_apply_kernel<<<gridElem, 256, 0, stream>>>(agg, ss, h, n4);
        hin = h;
    }

    float* mu = (float*)d_out;
    float* lv = (float*)d_out + (size_t)ND;

    // mu head
    gemm64_relu_kernel<<<gridGemm, 128, 0, stream>>>(h, muW, mub, tmp, N);
    hipMemsetAsync(stats, 0, 128 * sizeof(float), stream);
    bn_stats_kernel<<<1024, 256, 0, stream>>>(tmp, stats, N);
    bn_finalize_kernel<<<1, 64, 0, stream>>>(stats, mug, mubt, ss, invN);
    bn_apply_kernel<<<gridElem, 256, 0, stream>>>(tmp, ss, mu, n4);

    // logvar head
    gemm64_relu_kernel<<<gridGemm, 128, 0, stream>>>(h, lvW, lvb, tmp, N);
    hipMemsetAsync(stats, 0, 128 * sizeof(float), stream);
    bn_stats_kernel<<<1024, 256, 0, stream>>>(tmp, stats, N);
    bn_finalize_kernel<<<1, 64, 0, stream>>>(stats, lvg, lvbt, ss, invN);
    bn_apply_kernel<<<gridElem, 256, 0, stream>>>(tmp, ss, lv, n4);
}